// HybridQuantumNATModel_65481071398219
// MI455X (gfx1250) — compile-verified
//
#include <hip/hip_runtime.h>
#include <math.h>

typedef float v2f __attribute__((ext_vector_type(2)));
typedef float v8f __attribute__((ext_vector_type(8)));

// workspace float layout
#define WS_UR   0      // 256 floats: Re(U[i][j]) at i*16+j
#define WS_UI   256    // 256 floats: Im(U)
#define WS_WPIX 512    // 2304 floats: Wpix[p][w], p in [0,576)
// doubles (sum[4], sumsq[4]) live at byte offset 16384

__device__ inline v8f wmma4(v2f a, v2f b, v8f c) {
    return __builtin_amdgcn_wmma_f32_16x16x4_f32(false, a, false, b, (short)0, c,
                                                 false, false);
}

// ---------------- prep: build U (16x16 complex) and Wpix, zero BN sums -------
__global__ void qnat_prep(const float* __restrict__ pre_fc_w,
                          const float* __restrict__ weights,
                          float* __restrict__ wsf, double* __restrict__ sums) {
    int tid = threadIdx.x;               // 64 threads, 1 block
    if (tid < 8) sums[tid] = 0.0;
    for (int p = tid; p < 576; p += 64) {
        int r = p / 24, cc = p % 24;
        int k = (r / 6) * 4 + (cc / 6);
        for (int w = 0; w < 4; ++w)
            wsf[WS_WPIX + p * 4 + w] = pre_fc_w[w * 16 + k] * (1.0f / 36.0f);
    }
    if (tid < 16) {                      // thread j owns column j of U
        int j = tid;
        float ur[16], ui[16];
        for (int i = 0; i < 16; ++i) { ur[i] = (i == j) ? 1.f : 0.f; ui[i] = 0.f; }
        for (int l = 0; l < 3; ++l) {
            for (int w = 0; w < 4; ++w) {
                int bit = 8 >> w;        // wire w = bit (3-w) of state index
                // RX(weights[l][w][0]):  [[c,-is],[-is,c]]
                float th = weights[(l * 4 + w) * 2 + 0] * 0.5f;
                float c = cosf(th), s = sinf(th);
                for (int r0 = 0; r0 < 16; ++r0) {
                    if (r0 & bit) continue;
                    int r1 = r0 | bit;
                    float are = ur[r0], aim = ui[r0], bre = ur[r1], bim = ui[r1];
                    ur[r0] = c * are + s * bim;  ui[r0] = c * aim - s * bre;
                    ur[r1] = c * bre + s * aim;  ui[r1] = c * bim - s * are;
                }
                // RZ(weights[l][w][1]): diag(e^{-it/2}, e^{+it/2})
                th = weights[(l * 4 + w) * 2 + 1] * 0.5f;
                c = cosf(th); s = sinf(th);
                for (int r = 0; r < 16; ++r) {
                    float xx = ur[r], yy = ui[r];
                    if (r & bit) { ur[r] = xx * c - yy * s; ui[r] = yy * c + xx * s; }
                    else         { ur[r] = xx * c + yy * s; ui[r] = yy * c - xx * s; }
                }
            }
            for (int w = 0; w < 4; ++w) { // CNOT ring: swap rows (r, r|tb) where cb set
                int cb = 8 >> w, tb = 8 >> ((w + 1) & 3);
                for (int r = 0; r < 16; ++r) {
                    if ((r & cb) && !(r & tb)) {
                        int r2 = r | tb; float t;
                        t = ur[r]; ur[r] = ur[r2]; ur[r2] = t;
                        t = ui[r]; ui[r] = ui[r2]; ui[r2] = t;
                    }
                }
            }
        }
        for (int i = 0; i < 16; ++i) {
            wsf[WS_UR + i * 16 + j] = ur[i];
            wsf[WS_UI + i * 16 + j] = ui[i];
        }
    }
}

// ---------------- main: per-wave tile of 16 samples ---------------------------
__global__ __launch_bounds__(128) void qnat_main(
    const float* __restrict__ x, const float* __restrict__ pre_fc_b,
    const float* __restrict__ fc_w, const float* __restrict__ fc_b,
    const float* __restrict__ wsf, double* __restrict__ sums,
    float* __restrict__ out) {
    __shared__ float sWpix[2304];
    __shared__ float sStage[4][1792];   // per wave: 16 samples x 4 rows x 28 cols
    __shared__ float sCS[4][128];       // per wave: 16 samples x {c0..c3,s0..s3}
    __shared__ float sZ[4][64];         // per wave: 16 samples x 4 zvals

    const int tid  = threadIdx.x;
    const int wave = tid >> 5;
    const int lane = tid & 31;
    const int n    = lane & 15;         // matrix row/col index (M or N)
    const int h    = lane >> 4;         // lane half

    for (int i = tid; i < 2304; i += 128) sWpix[i] = wsf[WS_WPIX + i];
    __syncthreads();

    // Preload B-fragments of U^T (B[k=j][n=i] = U[i][j]); K = 4t + v + 2h
    v2f BUr[4], BUi[4];
#pragma unroll
    for (int t = 0; t < 4; ++t) {
        int j0 = 4 * t + 2 * h;
        BUr[t][0] = wsf[WS_UR + n * 16 + j0];  BUr[t][1] = wsf[WS_UR + n * 16 + j0 + 1];
        BUi[t][0] = wsf[WS_UI + n * 16 + j0];  BUi[t][1] = wsf[WS_UI + n * 16 + j0 + 1];
    }

    const int tileIdx = blockIdx.x * 4 + wave;
    const int s0 = tileIdx * 16;
    float* stage = sStage[wave];

    // ---- GEMM 1: encoded[16x4] = X_tile[16x576] @ Wpix[576x4] (K in 4s) ----
    v8f acc = {0.f, 0.f, 0.f, 0.f, 0.f, 0.f, 0.f, 0.f};
    for (int ch = 0; ch < 6; ++ch) {
        const int r0 = ch * 4;
        // stage 16 samples x rows r0..r0+3 (full 28 cols), coalesced float4
#pragma unroll
        for (int jj = 0; jj < 14; ++jj) {
            int idx = jj * 32 + lane;            // 0..447
            int ss  = idx / 28;
            int rem = idx - ss * 28;
            int rr  = rem / 7;
            int v4  = rem - rr * 7;
            float4 val = *(const float4*)(x + (size_t)(s0 + ss) * 784 +
                                          (size_t)(r0 + rr) * 28 + v4 * 4);
            *(float4*)(stage + (ss * 4 + rr) * 28 + v4 * 4) = val;
        }
        asm volatile("s_wait_dscnt 0" ::: "memory");
#pragma unroll
        for (int t = 0; t < 24; ++t) {
            int p0  = ch * 96 + 4 * t + 2 * h;   // pixel of A element v=0
            int pl  = p0 - ch * 96;
            int rl  = pl / 24, col = pl - rl * 24;   // col even -> pair in-row
            float2 av = *(const float2*)(stage + (n * 4 + rl) * 28 + col);
            v2f a; a[0] = av.x; a[1] = av.y;
            float b0 = sWpix[p0 * 4 + (n & 3)];
            float b1 = sWpix[(p0 + 1) * 4 + (n & 3)];
            v2f b; b[0] = (n < 4) ? b0 : 0.f; b[1] = (n < 4) ? b1 : 0.f;
            acc = wmma4(a, b, acc);
        }
        asm volatile("s_wait_dscnt 0" ::: "memory");
    }

    // ---- angles -> cos/sin, exchange via LDS ----
    float* cs = sCS[wave];
    {
        float bias = pre_fc_b[n & 3];
        if (n < 4) {
#pragma unroll
            for (int d = 0; d < 8; ++d) {
                float e = (acc[d] + bias) * 0.5f;
                int m = d + 8 * h;
                cs[m * 8 + n]     = cosf(e);
                cs[m * 8 + 4 + n] = sinf(e);
            }
        }
    }
    asm volatile("s_wait_dscnt 0" ::: "memory");

    // ---- build psi0 A-fragments: psi0[m][j] = prod_w (bit? -i s_w : c_w) ----
    float cw[4], sw[4];
#pragma unroll
    for (int w = 0; w < 4; ++w) { cw[w] = cs[n * 8 + w]; sw[w] = cs[n * 8 + 4 + w]; }
    v2f Ar[4], Ai[4];
#pragma unroll
    for (int t = 0; t < 4; ++t) {
#pragma unroll
        for (int v = 0; v < 2; ++v) {
            int j = 4 * t + v + 2 * h;
            float mag = 1.f;
#pragma unroll
            for (int w = 0; w < 4; ++w)
                mag *= ((j >> (3 - w)) & 1) ? sw[w] : cw[w];
            int p = __popc(j) & 3;               // (-i)^p
            Ar[t][v] = (p == 0) ? mag : (p == 2) ? -mag : 0.f;
            Ai[t][v] = (p == 1) ? -mag : (p == 3) ? mag : 0.f;
        }
    }

    // ---- GEMM 2: state = psi0 @ U^T (complex via 16 fp32 WMMAs) ----
    v8f accRe = {0.f,0.f,0.f,0.f,0.f,0.f,0.f,0.f};
    v8f accT  = {0.f,0.f,0.f,0.f,0.f,0.f,0.f,0.f};
    v8f accIm = {0.f,0.f,0.f,0.f,0.f,0.f,0.f,0.f};
#pragma unroll
    for (int t = 0; t < 4; ++t) accRe = wmma4(Ar[t], BUr[t], accRe);
#pragma unroll
    for (int t = 0; t < 4; ++t) accT  = wmma4(Ai[t], BUi[t], accT);
#pragma unroll
    for (int t = 0; t < 4; ++t) accIm = wmma4(Ar[t], BUi[t], accIm);
#pragma unroll
    for (int t = 0; t < 4; ++t) accIm = wmma4(Ai[t], BUr[t], accIm);

    v8f probs;
#pragma unroll
    for (int d = 0; d < 8; ++d) {
        float re = accRe[d] - accT[d];
        probs[d] = re * re + accIm[d] * accIm[d];
    }

    // ---- <Z_w> via Walsh-Hadamard over state index (lanes) ----
#pragma unroll
    for (int st = 1; st <= 8; st <<= 1) {
#pragma unroll
        for (int d = 0; d < 8; ++d) {
            float o = __shfl_xor(probs[d], st);
            probs[d] = (lane & st) ? (o - probs[d]) : (probs[d] + o);
        }
    }
    // zval for wire w sits in lane with n == (8 >> w)
    float* zs = sZ[wave];
    {
        int w = (n == 8) ? 0 : (n == 4) ? 1 : (n == 2) ? 2 : (n == 1) ? 3 : -1;
        if (w >= 0)
#pragma unroll
            for (int d = 0; d < 8; ++d) zs[(d + 8 * h) * 4 + w] = probs[d];
    }
    asm volatile("s_wait_dscnt 0" ::: "memory");

    // ---- logits + partial BN sums ----
    if (lane < 16) {
        int m = lane;
        float q0 = zs[m * 4 + 0], q1 = zs[m * 4 + 1];
        float q2 = zs[m * 4 + 2], q3 = zs[m * 4 + 3];
        float lg[4];
#pragma unroll
        for (int c = 0; c < 4; ++c)
            lg[c] = fc_b[c] + q0 * fc_w[c * 4 + 0] + q1 * fc_w[c * 4 + 1] +
                    q2 * fc_w[c * 4 + 2] + q3 * fc_w[c * 4 + 3];
        *(float4*)(out + (size_t)(s0 + m) * 4) =
            make_float4(lg[0], lg[1], lg[2], lg[3]);
        float ps[8];
#pragma unroll
        for (int c = 0; c < 4; ++c) { ps[c] = lg[c]; ps[4 + c] = lg[c] * lg[c]; }
#pragma unroll
        for (int k = 0; k < 8; ++k) {
            float v = ps[k];
            v += __shfl_xor(v, 1); v += __shfl_xor(v, 2);
            v += __shfl_xor(v, 4); v += __shfl_xor(v, 8);
            if (lane == 0) atomicAdd(&sums[k], (double)v);
        }
    }
}

// ---------------- BN finalize (in place on d_out) ----------------------------
__global__ void qnat_bn(float* __restrict__ out, const double* __restrict__ sums,
                        const float* __restrict__ gamma,
                        const float* __restrict__ beta, int B) {
    int s = blockIdx.x * blockDim.x + threadIdx.x;
    if (s >= B) return;
    float4 lg = *(const float4*)(out + (size_t)s * 4);
    float r[4] = {lg.x, lg.y, lg.z, lg.w};
    float o[4];
#pragma unroll
    for (int c = 0; c < 4; ++c) {
        float mean = (float)(sums[c] / (double)B);
        float var  = (float)(sums[4 + c] / (double)B) - mean * mean;
        float inv  = rsqrtf(var + 1e-5f);
        o[c] = (r[c] - mean) * inv * gamma[c] + beta[c];
    }
    *(float4*)(out + (size_t)s * 4) = make_float4(o[0], o[1], o[2], o[3]);
}

extern "C" void kernel_launch(void* const* d_in, const int* in_sizes, int n_in,
                              void* d_out, int out_size, void* d_ws, size_t ws_size,
                              hipStream_t stream) {
    const float* x        = (const float*)d_in[0];
    const float* pre_fc_w = (const float*)d_in[1];
    const float* pre_fc_b = (const float*)d_in[2];
    const float* weights  = (const float*)d_in[3];
    const float* fc_w     = (const float*)d_in[4];
    const float* fc_b     = (const float*)d_in[5];
    const float* gamma    = (const float*)d_in[6];
    const float* beta     = (const float*)d_in[7];
    float*  out  = (float*)d_out;
    float*  wsf  = (float*)d_ws;
    double* sums = (double*)((char*)d_ws + 16384);

    const int B = in_sizes[0] / 784;          // 65536
    hipLaunchKernelGGL(qnat_prep, dim3(1), dim3(64), 0, stream,
                       pre_fc_w, weights, wsf, sums);
    const int blocks = (B / 16) / 4;          // 4 waves/block, 16 samples/wave
    hipLaunchKernelGGL(qnat_main, dim3(blocks), dim3(128), 0, stream,
                       x, pre_fc_b, fc_w, fc_b, wsf, sums, out);
    hipLaunchKernelGGL(qnat_bn, dim3((B + 255) / 256), dim3(256), 0, stream,
                       out, sums, gamma, beta, B);
}